// DynamicPosition_79817672229414
// MI455X (gfx1250) — compile-verified
//
#include <hip/hip_runtime.h>
#include <hip/hip_bf16.h>

// ---- problem constants -----------------------------------------------------
#define LL      512
#define BB      8
#define NH      12
#define HD      64
#define HID     768
#define NBH     (BB * NH)            // 96
#define NBUCK   63                   // 2*BUCKET_SIZE-1
#define TPAD    64                   // padded bucket dim for tables
#define LUT_N   2048                 // rel in [-1024, 1023]
#define SCALE_F 0.07216878364870323f // 1/sqrt(192)

typedef __attribute__((ext_vector_type(2))) float v2f;
typedef __attribute__((ext_vector_type(4))) float v4f;
typedef __attribute__((ext_vector_type(8))) float v8f;

// ---------------------------------------------------------------------------
// Kernel 1: soft = sigmoid(hidden @ pos_w^T + pos_b)*1.2-0.1 ; cum = scan(soft)
// Also block 0 builds the log-bucket LUT (rel -> bucket index in [0,62]).
// grid = 96 (b*NH+h), block = 512 (one thread per l)
// ---------------------------------------------------------------------------
__global__ void __launch_bounds__(LL)
soft_cumsum_lut_kernel(const float* __restrict__ hidden,
                       const float* __restrict__ pos_w,
                       const float* __restrict__ pos_b,
                       float* __restrict__ cum,   // [96, 512]
                       int*   __restrict__ lut)   // [2048]
{
    __shared__ float s[LL];
    const int bh = blockIdx.x;
    const int b  = bh / NH;
    const int h  = bh % NH;
    const int t  = threadIdx.x;           // = l

    // dot(hidden[l,b,:], pos_w[h,:]), vectorized float4
    const float4* hv = (const float4*)(hidden + (size_t)(t * BB + b) * HID);
    const float4* wv = (const float4*)(pos_w + (size_t)h * HID);
    float acc = 0.0f;
    #pragma unroll 4
    for (int c = 0; c < HID / 4; ++c) {
        float4 a = hv[c];
        float4 w = wv[c];
        acc += a.x * w.x + a.y * w.y + a.z * w.z + a.w * w.w;
    }
    acc += pos_b[h];
    float soft = 1.0f / (1.0f + expf(-acc));
    soft = soft * 1.2f - 0.1f;

    // inclusive Hillis-Steele scan over 512 values in LDS
    s[t] = soft;
    __syncthreads();
    #pragma unroll
    for (int off = 1; off < LL; off <<= 1) {
        float v = (t >= off) ? s[t - off] : 0.0f;
        __syncthreads();
        s[t] += v;
        __syncthreads();
    }
    cum[(size_t)bh * LL + t] = s[t];

    // block 0 builds bucket LUT (transcendentals done once, not per element)
    if (bh == 0) {
        const float log_scale = 15.0f / logf(511.0f / 16.0f);
        #pragma unroll
        for (int k = 0; k < 4; ++k) {
            int rel = (t * 4 + k) - 1024;
            int sgn = (rel > 0) - (rel < 0);
            int abs_pos = (rel < 16 && rel > -16) ? 15 : min(abs(rel), 511);
            int bucket;
            if (abs_pos <= 16) {
                bucket = rel;
            } else {
                int log_pos = (int)ceilf(logf((float)abs_pos * (1.0f / 16.0f)) * log_scale) + 16;
                bucket = log_pos * sgn;
            }
            lut[t * 4 + k] = bucket + 31;   // in [0, 62]
        }
    }
}

// ---------------------------------------------------------------------------
// Kernel 2: bias tables via V_WMMA_F32_16X16X4_F32.
//  z==0: qb [bh,512,64]  : qb[bh,q,n]  = sum_d query[q,b,h,d] * k_rel[n,h,d]*SCALE
//  z==1: kbt[bh,512,64]  : kbt[bh,k,n] = sum_d key[k,b,h,d]   * q_rel[n,h,d]*SCALE
// grid = (32 Mtiles, 96 bh, 2), block = 128 (4 waves, one 16x16 tile each)
// ---------------------------------------------------------------------------
__global__ void __launch_bounds__(128)
bias_table_wmma_kernel(const float* __restrict__ query,
                       const float* __restrict__ key,
                       const float* __restrict__ k_rel,
                       const float* __restrict__ q_rel,
                       float* __restrict__ qb,
                       float* __restrict__ kbt)
{
    const int which = blockIdx.z;
    const float* src = which ? key   : query;   // [L,B,NH,HD]
    const float* rel = which ? q_rel : k_rel;   // [63,NH,HD]
    float*       out = which ? kbt   : qb;      // [96,512,64]

    const int bh    = blockIdx.y;
    const int b     = bh / NH;
    const int h     = bh % NH;
    const int Mbase = blockIdx.x * 16;

    const int wave  = threadIdx.x >> 5;
    const int lane  = threadIdx.x & 31;
    const int m     = lane & 15;        // row within tile (A) / col within tile (B,D)
    const int khalf = lane >> 4;        // selects K pair {0,1} vs {2,3}

    const int n  = wave * 16 + m;       // table column (bucket index)
    const float bscale = (n < NBUCK) ? SCALE_F : 0.0f;   // zero-pad col 63
    const int nIdx = min(n, NBUCK - 1);

    const float* srcRow = src + (((size_t)(Mbase + m) * BB + b) * NH + h) * HD;
    const float* relRow = rel + ((size_t)nIdx * NH + h) * HD;

    v8f acc = {0.f, 0.f, 0.f, 0.f, 0.f, 0.f, 0.f, 0.f};
    #pragma unroll
    for (int k0 = 0; k0 < HD; k0 += 4) {
        const int kk = k0 + 2 * khalf;
        v2f a, bf;
        a.x  = srcRow[kk];
        a.y  = srcRow[kk + 1];
        bf.x = relRow[kk]     * bscale;
        bf.y = relRow[kk + 1] * bscale;
        acc = __builtin_amdgcn_wmma_f32_16x16x4_f32(
                  /*neg_a=*/false, a, /*neg_b=*/false, bf,
                  /*c_mod=*/(short)0, acc, /*reuse_a=*/false, /*reuse_b=*/false);
    }

    // D layout: VGPR r -> (M = r + 8*khalf, N = lane&15) within tile
    float* orow = out + (size_t)bh * LL * TPAD;
    const int rowBase = Mbase + khalf * 8;
    #pragma unroll
    for (int r = 0; r < 8; ++r) {
        orow[(size_t)(rowBase + r) * TPAD + n] = acc[r];
    }
}

// ---------------------------------------------------------------------------
// Kernel 3 (bandwidth-bound): gather + lerp + store both outputs.
// 201 MB of write-once output vs 25 MB of L2-resident gather tables:
// use non-temporal (TH=NT) stores so the output stream does not evict the
// tables from L2 (192 MB < 201 MB of streamed writes otherwise cycles L2).
// grid = 96*512 (one block per (bh,i)), block = 128 (4 j's per thread)
// ---------------------------------------------------------------------------
__global__ void __launch_bounds__(128)
gather_out_kernel(const float* __restrict__ cum,   // [96,512]
                  const int*   __restrict__ lut,   // [2048] (L1-hot, 8KB)
                  const float* __restrict__ qb,    // [96,512,64]
                  const float* __restrict__ kbt,   // [96,512,64]
                  float* __restrict__ out)         // qpb[96,512,512] ++ kpb[...]
{
    __shared__ float qrow[TPAD];
    const int gid = blockIdx.x;        // bh*512 + i
    const int i   = gid & (LL - 1);
    const int bh  = gid >> 9;
    const int t   = threadIdx.x;

    if (t < TPAD) qrow[t] = qb[(size_t)gid * TPAD + t];
    const float ci = cum[gid];
    __syncthreads();

    const float4* cum4 = (const float4*)(cum + (size_t)bh * LL);
    const float4 cj4 = cum4[t];
    const int j0 = t * 4;
    const float* kr = kbt + (size_t)bh * LL * TPAD;

    float o1[4], o2[4];
    #pragma unroll
    for (int u = 0; u < 4; ++u) {
        const int j = j0 + u;
        const float cj = (&cj4.x)[u];
        const float sb = (j > i) ? 1.0f : ((j < i) ? -1.0f : 0.0f);
        const float idx = cj - ci + sb;
        const float f = floorf(idx);
        const float delta = idx - f;
        int fi = (int)f;
        fi = max(-1024, min(1023, fi));
        const int fb = lut[fi + 1024];
        const int cb = lut[min(fi + 1, 1023) + 1024];  // ==ceil bucket unless delta==0 (then unused)
        const float fq = qrow[fb];
        const float cq = qrow[cb];
        const float* krow = kr + (size_t)j * TPAD;
        const float fk = krow[fb];
        const float ck = krow[cb];
        o1[u] = fq + delta * (cq - fq);
        o2[u] = fk + delta * (ck - fk);
    }

    const size_t obase = (size_t)gid * LL + j0;
    v4f r1 = { o1[0], o1[1], o1[2], o1[3] };
    v4f r2 = { o2[0], o2[1], o2[2], o2[3] };
    __builtin_nontemporal_store(r1, (v4f*)(out + obase));
    __builtin_nontemporal_store(r2, (v4f*)(out + (size_t)NBH * LL * LL + obase));
}

// ---------------------------------------------------------------------------
extern "C" void kernel_launch(void* const* d_in, const int* in_sizes, int n_in,
                              void* d_out, int out_size, void* d_ws, size_t ws_size,
                              hipStream_t stream) {
    const float* hidden = (const float*)d_in[0];  // [512,8,768]
    const float* query  = (const float*)d_in[1];  // [512,8,12,64]
    const float* key    = (const float*)d_in[2];  // [512,8,12,64]
    const float* pos_w  = (const float*)d_in[3];  // [12,768]
    const float* pos_b  = (const float*)d_in[4];  // [12]
    const float* q_rel  = (const float*)d_in[5];  // [63,12,64]
    const float* k_rel  = (const float*)d_in[6];  // [63,12,64]

    // workspace layout (floats): cum[96*512] | lut[2048](int) | qb | kbt
    float* ws  = (float*)d_ws;
    float* cum = ws;
    int*   lut = (int*)(ws + (size_t)NBH * LL);
    float* qb  = ws + (size_t)NBH * LL + LUT_N;
    float* kbt = qb + (size_t)NBH * LL * TPAD;

    soft_cumsum_lut_kernel<<<NBH, LL, 0, stream>>>(hidden, pos_w, pos_b, cum, lut);

    dim3 g2(LL / 16, NBH, 2);
    bias_table_wmma_kernel<<<g2, 128, 0, stream>>>(query, key, k_rel, q_rel, qb, kbt);

    gather_out_kernel<<<NBH * LL, 128, 0, stream>>>(cum, lut, qb, kbt, (float*)d_out);
}